// GCN_32160715112515
// MI455X (gfx1250) — compile-verified
//
#include <hip/hip_runtime.h>
#include <hip/hip_bf16.h>
#include <math.h>

typedef __attribute__((ext_vector_type(2))) float v2f;
typedef __attribute__((ext_vector_type(8))) float v8f;

// ---------------------------------------------------------------------------
// small utility kernels
// ---------------------------------------------------------------------------
__global__ void k_zero_u32(unsigned* p, int n) {
    int i = blockIdx.x * blockDim.x + threadIdx.x;
    if (i < n) p[i] = 0u;
}

// count in-degree (real edges only; self loop added analytically)
__global__ void k_count(const int* __restrict__ dst, unsigned* __restrict__ cnt, int e) {
    int i = blockIdx.x * blockDim.x + threadIdx.x;
    if (i < e) atomicAdd(&cnt[dst[i]], 1u);
}

__global__ void k_dinv(const unsigned* __restrict__ cnt, float* __restrict__ dinv, int n) {
    int i = blockIdx.x * blockDim.x + threadIdx.x;
    if (i < n) dinv[i] = rsqrtf((float)(cnt[i] + 1u));   // +1 = self loop
}

// ---------------------------------------------------------------------------
// 3-kernel exclusive scan over counts -> CSR row offsets  (N <= 128*1024)
// ---------------------------------------------------------------------------
__global__ __launch_bounds__(256) void k_scan1(const unsigned* __restrict__ in,
                                               unsigned* __restrict__ out,
                                               unsigned* __restrict__ partials, int n) {
    __shared__ unsigned sh[256];
    int tid = threadIdx.x;
    int base = blockIdx.x * 1024 + tid * 4;
    unsigned v0 = (base + 0 < n) ? in[base + 0] : 0u;
    unsigned v1 = (base + 1 < n) ? in[base + 1] : 0u;
    unsigned v2 = (base + 2 < n) ? in[base + 2] : 0u;
    unsigned v3 = (base + 3 < n) ? in[base + 3] : 0u;
    unsigned s1 = v0, s2 = v0 + v1, s3 = v0 + v1 + v2, tsum = s3 + v3;
    sh[tid] = tsum;
    __syncthreads();
    for (int off = 1; off < 256; off <<= 1) {
        unsigned x = (tid >= off) ? sh[tid - off] : 0u;
        __syncthreads();
        sh[tid] += x;
        __syncthreads();
    }
    unsigned ex = sh[tid] - tsum;              // exclusive prefix for this thread
    if (base + 0 < n) out[base + 0] = ex;
    if (base + 1 < n) out[base + 1] = ex + s1;
    if (base + 2 < n) out[base + 2] = ex + s2;
    if (base + 3 < n) out[base + 3] = ex + s3;
    if (tid == 255) partials[blockIdx.x] = sh[255];
}

__global__ __launch_bounds__(128) void k_scan2(unsigned* __restrict__ partials, int nb) {
    __shared__ unsigned sh[128];
    int tid = threadIdx.x;
    unsigned v = (tid < nb) ? partials[tid] : 0u;
    sh[tid] = v;
    __syncthreads();
    for (int off = 1; off < 128; off <<= 1) {
        unsigned x = (tid >= off) ? sh[tid - off] : 0u;
        __syncthreads();
        sh[tid] += x;
        __syncthreads();
    }
    partials[tid] = sh[tid] - v;               // exclusive
}

__global__ void k_scan3(unsigned* __restrict__ row, const unsigned* __restrict__ partials,
                        unsigned* __restrict__ cursor, int n, int e) {
    int i = blockIdx.x * blockDim.x + threadIdx.x;
    if (i < n) {
        unsigned r = row[i] + partials[i >> 10];
        row[i] = r;
        cursor[i] = r;
    }
    if (i == 0) row[n] = (unsigned)e;          // total edges
}

// fill CSR column (source node per slot)
__global__ void k_fill(const int* __restrict__ src, const int* __restrict__ dst,
                       unsigned* __restrict__ cursor, unsigned* __restrict__ col, int e) {
    int i = blockIdx.x * blockDim.x + threadIdx.x;
    if (i < e) {
        unsigned slot = atomicAdd(&cursor[dst[i]], 1u);
        col[slot] = (unsigned)src[i];
    }
}

// ---------------------------------------------------------------------------
// pack W1 [256x128] into WMMA B-fragment lane order:
// Bp[((kstep*8 + ntile)*32 + lane)*2 + {0,1}]
// lane<16  -> (K=k0,  N=nt*16+lane), (K=k0+1, N)
// lane>=16 -> (K=k0+2,N=nt*16+lane-16), (K=k0+3, N)
// ---------------------------------------------------------------------------
__global__ void k_packB(const float* __restrict__ W1, float* __restrict__ Bp) {
    int t = blockIdx.x * blockDim.x + threadIdx.x;
    if (t < 64 * 8 * 32) {
        int kstep = t >> 8;
        int nt = (t >> 5) & 7;
        int lane = t & 31;
        int k0 = kstep * 4 + ((lane >> 4) << 1);
        int nn = nt * 16 + (lane & 15);
        Bp[2 * t + 0] = W1[k0 * 128 + nn];
        Bp[2 * t + 1] = W1[(k0 + 1) * 128 + nn];
    }
}

// ---------------------------------------------------------------------------
// GEMM1: H1[N,128] = X[N,256] @ W1[256,128] using V_WMMA_F32_16X16X4_F32.
// One block = 32 M-rows staged in LDS; 8 waves each own one 16-wide N tile
// and run TWO independent accumulator chains (M-subtiles 0-15 and 16-31)
// sharing each B fragment -> 2x WMMA ILP, 2x B-traffic amortization.
// Requires N % 32 == 0 (true: 100000 = 3125*32).
// ---------------------------------------------------------------------------
__global__ __launch_bounds__(256) void k_gemm1(const float* __restrict__ X,
                                               const float* __restrict__ Bp,
                                               float* __restrict__ H1) {
    __shared__ float tile[32 * 260];           // 32 rows, stride 260 (bank skew)
    const int tid = threadIdx.x;
    const int wave = tid >> 5;
    const int lane = tid & 31;
    const int mt = blockIdx.x;

    // cooperative load of 32x256 X tile (2048 float4, 8 per thread)
    const float4* Xf4 = (const float4*)(X + (size_t)mt * 32 * 256);
    float4* Tf4 = (float4*)tile;
    for (int i = 0; i < 8; ++i) {
        int idx = tid + 256 * i;               // 0..2047
        int r = idx >> 6;
        int c = idx & 63;
        Tf4[r * 65 + c] = Xf4[r * 64 + c];
    }
    __syncthreads();

    v8f acc0 = {};
    v8f acc1 = {};
    const int arow = lane & 15;
    const int coff = (lane >> 4) << 1;
    const v2f* Bpv = (const v2f*)Bp;
    for (int ks = 0; ks < 64; ++ks) {
        v2f a0 = *(const v2f*)&tile[arow * 260 + ks * 4 + coff];
        v2f a1 = *(const v2f*)&tile[(arow + 16) * 260 + ks * 4 + coff];
        v2f b = Bpv[(ks * 8 + wave) * 32 + lane];
        acc0 = __builtin_amdgcn_wmma_f32_16x16x4_f32(
            false, a0, false, b, (short)0, acc0, false, false);
        acc1 = __builtin_amdgcn_wmma_f32_16x16x4_f32(
            false, a1, false, b, (short)0, acc1, false, false);
    }

    // C/D layout: VGPR r -> lanes 0-15: (M=r, N=lane); lanes 16-31: (M=r+8, N=lane-16)
    const int mbase = mt * 32 + ((lane >> 4) << 3);
    const int ncol = wave * 16 + (lane & 15);
    for (int r = 0; r < 8; ++r) {
        H1[(mbase + r) * 128 + ncol] = acc0[r];
        H1[(mbase + 16 + r) * 128 + ncol] = acc1[r];
    }
}

// ---------------------------------------------------------------------------
// Aggregate layer 1: one wave per node, float4 per lane (128 features).
// out = dinv[n]*sum_s dinv[s]*H1[s] + dinv[n]^2*H1[n] + b1, then ReLU.
// ---------------------------------------------------------------------------
__global__ __launch_bounds__(256) void k_agg1(const unsigned* __restrict__ row,
                                              const unsigned* __restrict__ col,
                                              const float* __restrict__ dinv,
                                              const float* __restrict__ H1,
                                              const float* __restrict__ b1,
                                              float* __restrict__ A1, int n) {
    const int node = blockIdx.x * 8 + (threadIdx.x >> 5);
    const int lane = threadIdx.x & 31;
    if (node >= n) return;
    float ax = 0.f, ay = 0.f, az = 0.f, aw = 0.f;
    const unsigned beg = row[node], end = row[node + 1];
    for (unsigned i = beg; i < end; ++i) {
        const int s = (int)col[i];
        const float w = dinv[s];
        const float4 h = ((const float4*)(H1 + (size_t)s * 128))[lane];
        ax += w * h.x; ay += w * h.y; az += w * h.z; aw += w * h.w;
    }
    const float di = dinv[node];
    const float dii = di * di;
    const float4 hs = ((const float4*)(H1 + (size_t)node * 128))[lane];
    const float4 bb = ((const float4*)b1)[lane];
    float4 r;
    r.x = fmaxf(di * ax + dii * hs.x + bb.x, 0.f);
    r.y = fmaxf(di * ay + dii * hs.y + bb.y, 0.f);
    r.z = fmaxf(di * az + dii * hs.z + bb.z, 0.f);
    r.w = fmaxf(di * aw + dii * hs.w + bb.w, 0.f);
    ((float4*)(A1 + (size_t)node * 128))[lane] = r;
}

// ---------------------------------------------------------------------------
// GEMM2: H2[N,8] = A1[N,128] @ W2[128,8]; wave per node, K split across lanes.
// ---------------------------------------------------------------------------
__global__ __launch_bounds__(256) void k_gemm2(const float* __restrict__ A1,
                                               const float* __restrict__ W2,
                                               float* __restrict__ H2, int n) {
    __shared__ float w2s[128 * 8];
    const int tid = threadIdx.x;
    ((float4*)w2s)[tid] = ((const float4*)W2)[tid];   // 256*4 = 1024 floats
    __syncthreads();
    const int node = blockIdx.x * 8 + (tid >> 5);
    const int lane = tid & 31;
    if (node >= n) return;
    const float4 a = ((const float4*)(A1 + (size_t)node * 128))[lane]; // k = lane*4..+3
    float p[8];
    for (int o = 0; o < 8; ++o) p[o] = 0.f;
    const int kb = lane * 4;
    for (int o = 0; o < 8; ++o) {
        p[o] = a.x * w2s[(kb + 0) * 8 + o] + a.y * w2s[(kb + 1) * 8 + o] +
               a.z * w2s[(kb + 2) * 8 + o] + a.w * w2s[(kb + 3) * 8 + o];
    }
    for (int off = 16; off >= 1; off >>= 1)
        for (int o = 0; o < 8; ++o) p[o] += __shfl_xor(p[o], off, 32);
    if (lane == 0) {
        float4 r0 = make_float4(p[0], p[1], p[2], p[3]);
        float4 r1 = make_float4(p[4], p[5], p[6], p[7]);
        ((float4*)(H2 + (size_t)node * 8))[0] = r0;
        ((float4*)(H2 + (size_t)node * 8))[1] = r1;
    }
}

// ---------------------------------------------------------------------------
// Aggregate layer 2 + bias + log_softmax. 8 lanes per node (4 nodes / wave).
// ---------------------------------------------------------------------------
__global__ __launch_bounds__(256) void k_agg2(const unsigned* __restrict__ row,
                                              const unsigned* __restrict__ col,
                                              const float* __restrict__ dinv,
                                              const float* __restrict__ H2,
                                              const float* __restrict__ b2,
                                              float* __restrict__ out, int n) {
    const int tid = threadIdx.x;
    const int wave = tid >> 5;
    const int lane = tid & 31;
    const int g = lane >> 3;
    const int o = lane & 7;
    const int node = blockIdx.x * 32 + wave * 4 + g;
    if (node >= n) return;
    float acc = 0.f;
    const unsigned beg = row[node], end = row[node + 1];
    for (unsigned i = beg; i < end; ++i) {
        const int s = (int)col[i];
        acc += dinv[s] * H2[(size_t)s * 8 + o];
    }
    const float di = dinv[node];
    float v = di * acc + di * di * H2[(size_t)node * 8 + o] + b2[o];
    // log-softmax across the 8 lanes of this group
    float m = v;
    for (int off = 1; off < 8; off <<= 1) m = fmaxf(m, __shfl_xor(m, off, 8));
    float e = __expf(v - m);
    float s8 = e;
    for (int off = 1; off < 8; off <<= 1) s8 += __shfl_xor(s8, off, 8);
    out[(size_t)node * 8 + o] = v - m - __logf(s8);
}

// ---------------------------------------------------------------------------
// launcher
// ---------------------------------------------------------------------------
extern "C" void kernel_launch(void* const* d_in, const int* in_sizes, int n_in,
                              void* d_out, int out_size, void* d_ws, size_t ws_size,
                              hipStream_t stream) {
    const float* x  = (const float*)d_in[0];
    const int*  ei  = (const int*)d_in[1];     // [2,E]: row 0 = src, row 1 = dst
    const float* W1 = (const float*)d_in[2];
    const float* b1 = (const float*)d_in[3];
    const float* W2 = (const float*)d_in[4];
    const float* b2 = (const float*)d_in[5];
    float* out = (float*)d_out;

    const int N = in_sizes[0] / 256;
    const int E = in_sizes[1] / 2;
    const int* src = ei;
    const int* dst = ei + E;

    char* ws = (char*)d_ws;
    size_t off = 0;
    auto carve = [&](size_t bytes) -> void* {
        void* p = (void*)(ws + off);
        off += (bytes + 255) & ~(size_t)255;
        return p;
    };
    unsigned* count   = (unsigned*)carve((size_t)N * 4);
    unsigned* row     = (unsigned*)carve((size_t)(N + 1) * 4);
    unsigned* cursor  = (unsigned*)carve((size_t)N * 4);
    unsigned* partial = (unsigned*)carve(512);
    float*    dinv    = (float*)carve((size_t)N * 4);
    unsigned* col     = (unsigned*)carve((size_t)E * 4);
    float*    Bp      = (float*)carve((size_t)64 * 8 * 32 * 2 * 4);
    float*    H1      = (float*)carve((size_t)N * 128 * 4);
    float*    A1      = (float*)carve((size_t)N * 128 * 4);
    float*    H2      = (float*)carve((size_t)N * 8 * 4);

    const int TB = 256;
    const int gN  = (N + TB - 1) / TB;
    const int gE  = (E + TB - 1) / TB;
    const int nb1 = (N + 1023) / 1024;         // <= 128

    // degree + CSR build
    k_zero_u32<<<gN, TB, 0, stream>>>(count, N);
    k_count<<<gE, TB, 0, stream>>>(dst, count, E);
    k_scan1<<<nb1, TB, 0, stream>>>(count, row, partial, N);
    k_scan2<<<1, 128, 0, stream>>>(partial, nb1);
    k_scan3<<<gN, TB, 0, stream>>>(row, partial, cursor, N, E);
    k_dinv<<<gN, TB, 0, stream>>>(count, dinv, N);
    k_fill<<<gE, TB, 0, stream>>>(src, dst, cursor, col, E);

    // layer 1: WMMA GEMM then gather-aggregate (+bias, ReLU)
    k_packB<<<64, TB, 0, stream>>>(W1, Bp);
    k_gemm1<<<N / 32, TB, 0, stream>>>(x, Bp, H1);
    k_agg1<<<(N + 7) / 8, TB, 0, stream>>>(row, col, dinv, H1, b1, A1, N);

    // layer 2: small GEMM, gather-aggregate, log-softmax
    k_gemm2<<<(N + 7) / 8, TB, 0, stream>>>(A1, W2, H2, N);
    k_agg2<<<(N + 31) / 32, TB, 0, stream>>>(row, col, dinv, H2, b2, out, N);
}